// GraphAttention_79517024518470
// MI455X (gfx1250) — compile-verified
//
#include <hip/hip_runtime.h>
#include <math.h>

#define N_NODES  50000
#define DIM      128
#define HEADS    8
#define HEAD_DIM 16
#define NEDGE    800000
#define EDGE_DIM 4

typedef __attribute__((ext_vector_type(2))) float v2f;
typedef __attribute__((ext_vector_type(8))) float v8f;

// ---------------------------------------------------------------------------
// Wave-level 16x16 f32 tile of  A(16xDIM, LDS) @ W(DIMxDIM row-major cols c0..c0+15)
// using V_WMMA_F32_16X16X4_F32, K-loop of 32 MMAs.
// ---------------------------------------------------------------------------
__device__ __forceinline__ v8f wmma_xW(const float* __restrict__ Atile,
                                       const float* __restrict__ W, int c0) {
  const int lane = threadIdx.x & 31;
  const int half = lane >> 4;   // 0: K even-pair low, 1: K pair high
  const int l    = lane & 15;   // row (A) / col (B)
  v8f acc = {};
#pragma unroll
  for (int k0 = 0; k0 < DIM; k0 += 4) {
    v2f a, b;
    a.x = Atile[l * DIM + k0 + 2 * half + 0];
    a.y = Atile[l * DIM + k0 + 2 * half + 1];
    b.x = W[(k0 + 2 * half + 0) * DIM + c0 + l];
    b.y = W[(k0 + 2 * half + 1) * DIM + c0 + l];
    acc = __builtin_amdgcn_wmma_f32_16x16x4_f32(false, a, false, b,
                                                (short)0, acc, false, false);
  }
  return acc;
}

// ---------------------------------------------------------------------------
// Kernel 1: fused Q/K/V GEMMs. grid=(N/16, 3), block=256 (8 waves: one col-tile each)
// ---------------------------------------------------------------------------
__global__ void qkv_gemm(const float* __restrict__ x,
                         const float* __restrict__ Wq, const float* __restrict__ bq,
                         const float* __restrict__ Wk, const float* __restrict__ bk,
                         const float* __restrict__ Wv, const float* __restrict__ bv,
                         float* __restrict__ q, float* __restrict__ k,
                         float* __restrict__ v) {
  __shared__ float tileA[16 * DIM];
  const int r0 = blockIdx.x * 16;
  const float* W; const float* b; float* out;
  if (blockIdx.y == 0)      { W = Wq; b = bq; out = q; }
  else if (blockIdx.y == 1) { W = Wk; b = bk; out = k; }
  else                      { W = Wv; b = bv; out = v; }

  for (int i = threadIdx.x; i < 16 * DIM; i += blockDim.x)
    tileA[i] = x[(size_t)r0 * DIM + i];
  __syncthreads();

  const int c0 = (threadIdx.x >> 5) * 16;
  v8f acc = wmma_xW(tileA, W, c0);

  const int lane = threadIdx.x & 31;
  const int half = lane >> 4;
  const int l    = lane & 15;
  const float bb = b[c0 + l];
#pragma unroll
  for (int r = 0; r < 8; ++r)
    out[(size_t)(r0 + r + 8 * half) * DIM + c0 + l] = acc[r] + bb;
}

// ---------------------------------------------------------------------------
// Kernel 2: init scratch (max keys = -inf key, sums = 0, agg = 0)
// ---------------------------------------------------------------------------
__global__ void init_buffers(int* __restrict__ maxkey, float* __restrict__ sum,
                             float* __restrict__ agg) {
  const int t = blockIdx.x * blockDim.x + threadIdx.x;
  if (t < N_NODES * DIM) agg[t] = 0.0f;
  if (t < N_NODES * HEADS) { maxkey[t] = (int)0x80000000; sum[t] = 0.0f; }
}

// monotone float <-> ordered-int key
__device__ __forceinline__ int f2ord(float f) {
  int i = __float_as_int(f);
  return (i >= 0) ? i : (i ^ 0x7fffffff);
}
__device__ __forceinline__ float ord2f(int i) {
  return __int_as_float((i >= 0) ? i : (i ^ 0x7fffffff));
}

// ---------------------------------------------------------------------------
// Kernel 3: per-edge logits  attn[e,h] = 0.25*<q[dst,h],k[src,h]> + edge_attr@We + be
// ---------------------------------------------------------------------------
__global__ void edge_logits(const float* __restrict__ q, const float* __restrict__ k,
                            const int* __restrict__ src, const int* __restrict__ dst,
                            const float* __restrict__ edge_attr,
                            const float* __restrict__ We, const float* __restrict__ be,
                            float* __restrict__ attn) {
  const int t = blockIdx.x * blockDim.x + threadIdx.x;
  if (t >= NEDGE * HEADS) return;
  const int e = t >> 3, h = t & 7;
  const int s = src[e], d = dst[e];
  const float4* qp = (const float4*)(q + (size_t)d * DIM + h * HEAD_DIM);
  const float4* kp = (const float4*)(k + (size_t)s * DIM + h * HEAD_DIM);
  float acc = 0.0f;
#pragma unroll
  for (int i = 0; i < 4; ++i) {
    float4 a = qp[i], b = kp[i];
    acc += a.x * b.x + a.y * b.y + a.z * b.z + a.w * b.w;
  }
  acc *= 0.25f;  // HD^-0.5 = 1/sqrt(16)
  float eb = be[h];
#pragma unroll
  for (int j = 0; j < EDGE_DIM; ++j)
    eb += edge_attr[(size_t)e * EDGE_DIM + j] * We[j * HEADS + h];
  attn[t] = acc + eb;
}

// ---------------------------------------------------------------------------
// Kernel 4: segment max over dst (ordered-int atomicMax)
// ---------------------------------------------------------------------------
__global__ void seg_max(const float* __restrict__ attn, const int* __restrict__ dst,
                        int* __restrict__ maxkey) {
  const int t = blockIdx.x * blockDim.x + threadIdx.x;
  if (t >= NEDGE * HEADS) return;
  const int e = t >> 3, h = t & 7;
  atomicMax(&maxkey[dst[e] * HEADS + h], f2ord(attn[t]));
}

// ---------------------------------------------------------------------------
// Kernel 5: e = exp(attn - max[dst]); segment sum
// ---------------------------------------------------------------------------
__global__ void exp_sum(float* __restrict__ attn, const int* __restrict__ dst,
                        const int* __restrict__ maxkey, float* __restrict__ sum) {
  const int t = blockIdx.x * blockDim.x + threadIdx.x;
  if (t >= NEDGE * HEADS) return;
  const int e = t >> 3, h = t & 7;
  const int idx = dst[e] * HEADS + h;
  const float ex = __expf(attn[t] - ord2f(maxkey[idx]));
  attn[t] = ex;
  atomicAdd(&sum[idx], ex);
}

// ---------------------------------------------------------------------------
// Kernel 6: agg[dst,:] += (e/sum) * v[src,:]   (one thread per edge-channel)
// ---------------------------------------------------------------------------
__global__ void aggregate(const float* __restrict__ attn, const float* __restrict__ sum,
                          const float* __restrict__ v, const int* __restrict__ src,
                          const int* __restrict__ dst, float* __restrict__ agg) {
  const int t = blockIdx.x * blockDim.x + threadIdx.x;  // < E*DIM = 102.4M
  if (t >= NEDGE * DIM) return;
  const int e = t >> 7, d = t & 127, h = d >> 4;
  const int s = src[e], dd = dst[e];
  const float a = attn[e * HEADS + h] / fmaxf(sum[dd * HEADS + h], 1e-6f);
  atomicAdd(&agg[(size_t)dd * DIM + d], a * v[(size_t)s * DIM + d]);
}

// ---------------------------------------------------------------------------
// Kernel 7: out = GroupNorm8(agg @ Wo + bo) * gamma + beta, fused GN epilogue.
// One 16-col tile == one GroupNorm group; per-row stats via half-wave shfl_xor.
// ---------------------------------------------------------------------------
__global__ void out_gemm_gn(const float* __restrict__ agg,
                            const float* __restrict__ Wo, const float* __restrict__ bo,
                            const float* __restrict__ gamma, const float* __restrict__ beta,
                            float* __restrict__ out) {
  __shared__ float tileA[16 * DIM];
  const int r0 = blockIdx.x * 16;
  for (int i = threadIdx.x; i < 16 * DIM; i += blockDim.x)
    tileA[i] = agg[(size_t)r0 * DIM + i];
  __syncthreads();

  const int c0 = (threadIdx.x >> 5) * 16;
  v8f acc = wmma_xW(tileA, Wo, c0);

  const int lane = threadIdx.x & 31;
  const int half = lane >> 4;
  const int l    = lane & 15;
  const float bb = bo[c0 + l];
  const float gm = gamma[c0 + l];
  const float bt = beta[c0 + l];
#pragma unroll
  for (int r = 0; r < 8; ++r) {
    const float val = acc[r] + bb;
    float s1 = val, s2 = val * val;
#pragma unroll
    for (int m = 1; m < 16; m <<= 1) {   // reduce within 16-lane half (one row)
      s1 += __shfl_xor(s1, m, 32);
      s2 += __shfl_xor(s2, m, 32);
    }
    const float mu  = s1 * (1.0f / 16.0f);
    const float var = s2 * (1.0f / 16.0f) - mu * mu;
    const float inv = rsqrtf(var + 1e-5f);
    out[(size_t)(r0 + r + 8 * half) * DIM + c0 + l] = (val - mu) * inv * gm + bt;
  }
}

// ---------------------------------------------------------------------------
extern "C" void kernel_launch(void* const* d_in, const int* in_sizes, int n_in,
                              void* d_out, int out_size, void* d_ws, size_t ws_size,
                              hipStream_t stream) {
  const float* x        = (const float*)d_in[0];
  const int*   ei       = (const int*)d_in[1];
  const float* edge_att = (const float*)d_in[2];
  const float* Wq = (const float*)d_in[3];
  const float* bq = (const float*)d_in[4];
  const float* Wk = (const float*)d_in[5];
  const float* bk = (const float*)d_in[6];
  const float* Wv = (const float*)d_in[7];
  const float* bv = (const float*)d_in[8];
  const float* We = (const float*)d_in[9];
  const float* be = (const float*)d_in[10];
  const float* Wo = (const float*)d_in[11];
  const float* bo = (const float*)d_in[12];
  const float* gamma = (const float*)d_in[13];
  const float* beta  = (const float*)d_in[14];
  const int* src = ei;
  const int* dst = ei + NEDGE;

  const size_t ND = (size_t)N_NODES * DIM;     // 6.4M floats
  const size_t EH = (size_t)NEDGE * HEADS;     // 6.4M floats
  const size_t NH = (size_t)N_NODES * HEADS;   // 0.4M floats

  float* ws   = (float*)d_ws;
  float* q    = ws;
  float* k    = q + ND;
  float* v    = k + ND;
  float* attn = v + ND;
  int*   mkey = (int*)(attn + EH);
  float* sum  = (float*)(mkey + NH);
  float* agg  = sum + NH;

  const dim3 blk(256);
  qkv_gemm<<<dim3(N_NODES / 16, 3), blk, 0, stream>>>(x, Wq, bq, Wk, bk, Wv, bv, q, k, v);
  init_buffers<<<(N_NODES * DIM + 255) / 256, blk, 0, stream>>>(mkey, sum, agg);
  edge_logits<<<(NEDGE * HEADS + 255) / 256, blk, 0, stream>>>(q, k, src, dst, edge_att,
                                                               We, be, attn);
  seg_max<<<(NEDGE * HEADS + 255) / 256, blk, 0, stream>>>(attn, dst, mkey);
  exp_sum<<<(NEDGE * HEADS + 255) / 256, blk, 0, stream>>>(attn, dst, mkey, sum);
  aggregate<<<(NEDGE * DIM + 255) / 256, blk, 0, stream>>>(attn, sum, v, src, dst, agg);
  out_gemm_gn<<<N_NODES / 16, blk, 0, stream>>>(agg, Wo, bo, gamma, beta, (float*)d_out);
}